// PatchMerger_86131274154372
// MI455X (gfx1250) — compile-verified
//
#include <hip/hip_runtime.h>
#include <math.h>

typedef __attribute__((ext_vector_type(2))) float v2f;
typedef __attribute__((ext_vector_type(8))) float v8f;

#define B_ 16
#define N_ 4096
#define D_ 768
#define M_ 8
#define CHUNKS 8
#define CN (N_ / CHUNKS)            // 512 rows per chunk
#define NWAVES 4
#define BLOCK (NWAVES * 32)         // 128 threads, 4 wave32
#define TILES_PER_WAVE (CN / 16 / NWAVES)  // 8
#define EPS_ 1e-3f
#define NEG_BIG (-1e30f)

// LDS float layout:
//  sQ   [16][768]  (rows 8..15 zero-padded)
//  sG   [768], sBe [768]
//  sAcc [NWAVES][8][768]
//  sMu  [NWAVES][16], sRs [NWAVES][16]
//  sP   [NWAVES][16][16]
//  sWmax[NWAVES][8], sWl [NWAVES][8]
//  sBmax[8] (+8 pad)
#define LDS_FLOATS (16*D_ + 2*D_ + NWAVES*8*D_ + NWAVES*16*2 + NWAVES*256 + NWAVES*8*2 + 16)

extern "C" __global__ __launch_bounds__(BLOCK)
void pm_fused(const float* __restrict__ x, const float* __restrict__ q,
              const float* __restrict__ gamma, const float* __restrict__ beta,
              float* __restrict__ pmax, float* __restrict__ pl,
              float* __restrict__ pacc)
{
    extern __shared__ float lds[];
    float* sQ    = lds;                       // 16*768
    float* sG    = sQ + 16 * D_;              // 768
    float* sBe   = sG + D_;                   // 768
    float* sAcc  = sBe + D_;                  // NWAVES*8*768
    float* sMu   = sAcc + NWAVES * 8 * D_;    // NWAVES*16
    float* sRs   = sMu + NWAVES * 16;         // NWAVES*16
    float* sP    = sRs + NWAVES * 16;         // NWAVES*256
    float* sWmax = sP + NWAVES * 256;         // NWAVES*8
    float* sWl   = sWmax + NWAVES * 8;        // NWAVES*8
    float* sBmax = sWl + NWAVES * 8;          // 8 (+8 pad)

    const int tid  = threadIdx.x;
    const int wave = tid >> 5;
    const int lane = tid & 31;
    const int half = lane >> 4;     // 0: lanes 0-15, 1: lanes 16-31
    const int l16  = lane & 15;
    const int b     = blockIdx.x / CHUNKS;
    const int chunk = blockIdx.x % CHUNKS;
    const long xbase = (long)b * N_ * D_;

    // ---- stage Q (zero-padded to 16 rows), gamma, beta; zero accumulators
    for (int i = tid; i < 16 * D_; i += BLOCK) {
        int r = i / D_;
        sQ[i] = (r < M_) ? q[i] : 0.0f;
    }
    for (int i = tid; i < D_; i += BLOCK) { sG[i] = gamma[i]; sBe[i] = beta[i]; }
    for (int i = tid; i < NWAVES * 8 * D_; i += BLOCK) sAcc[i] = 0.0f;
    __syncthreads();

    float* myAcc = sAcc + wave * 8 * D_;
    float* myMu  = sMu + wave * 16;
    float* myRs  = sRs + wave * 16;
    float* myP   = sP + wave * 256;

    float rowmax[8], rowsum[8];
#pragma unroll
    for (int r = 0; r < 8; ++r) { rowmax[r] = NEG_BIG; rowsum[r] = 0.0f; }

    const float scale = 0.03608439182435161f;  // 768^-0.5

    for (int t = 0; t < TILES_PER_WAVE; ++t) {
        const int n0 = chunk * CN + (wave * TILES_PER_WAVE + t) * 16;

        // ---- pass 1: LayerNorm statistics for the 16 rows of this tile
        for (int r = 0; r < 16; ++r) {
            const float* row = x + xbase + (long)(n0 + r) * D_;
            float s = 0.0f, s2 = 0.0f;
#pragma unroll
            for (int i = 0; i < D_ / (32 * 4); ++i) {      // 6 x float4 per lane
                float4 v = *(const float4*)(row + lane * 4 + i * 128);
                s  += v.x + v.y + v.z + v.w;
                s2 += v.x * v.x + v.y * v.y + v.z * v.z + v.w * v.w;
            }
#pragma unroll
            for (int off = 16; off >= 1; off >>= 1) {
                s  += __shfl_xor(s, off, 32);
                s2 += __shfl_xor(s2, off, 32);
            }
            float mu  = s * (1.0f / D_);
            float var = s2 * (1.0f / D_) - mu * mu;
            if (lane == 0) { myMu[r] = mu; myRs[r] = rsqrtf(var + EPS_); }
        }

        // per-lane LN params for the B-fragment row it serves (row = l16)
        const float  mu_r = myMu[l16];
        const float  rs_r = myRs[l16];
        const float* xrow = x + xbase + (long)(n0 + l16) * D_;

        // ---- GEMM1: S[16,16] = Qpad[16,768] x Xn^T[768,16]  (fp32 WMMA, K=4 steps)
        v8f S = {0.f, 0.f, 0.f, 0.f, 0.f, 0.f, 0.f, 0.f};
#pragma unroll 4
        for (int k = 0; k < D_; k += 4) {
            const int kf = k + half * 2;                  // this lane's K pair
            v2f a;                                        // A: Q[M=l16][kf..kf+1]
            a.x = sQ[l16 * D_ + kf];
            a.y = sQ[l16 * D_ + kf + 1];
            float2 xv = *(const float2*)(xrow + kf);      // B: xn[N=l16][kf..kf+1]
            v2f bm;
            bm.x = (xv.x - mu_r) * rs_r * sG[kf]     + sBe[kf];
            bm.y = (xv.y - mu_r) * rs_r * sG[kf + 1] + sBe[kf + 1];
            S = __builtin_amdgcn_wmma_f32_16x16x4_f32(false, a, false, bm,
                                                      (short)0, S, false, false);
        }

        // ---- online softmax over this tile's 16 columns
        float corr[8];
#pragma unroll
        for (int r = 0; r < 8; ++r) {
            float sim = S[r] * scale;                     // row m=r (+8 for upper half)
            float tmax = sim;
#pragma unroll
            for (int off = 8; off >= 1; off >>= 1)
                tmax = fmaxf(tmax, __shfl_xor(tmax, off, 16));
            float nm = fmaxf(rowmax[r], tmax);
            float p  = __expf(sim - nm);
            float c  = __expf(rowmax[r] - nm);
            float ps = p;
#pragma unroll
            for (int off = 8; off >= 1; off >>= 1)
                ps += __shfl_xor(ps, off, 16);
            rowsum[r] = rowsum[r] * c + ps;
            rowmax[r] = nm;
            corr[r]   = c;
            myP[(r + half * 8) * 16 + l16] = p;           // P[m][n'] to LDS
        }

        // ---- GEMM2: D[16,16] = P[16,16] x Xn[16, fblk] per 16-wide feature block
        for (int f0 = 0; f0 < D_; f0 += 16) {
            const int   f  = f0 + l16;
            const float g  = sG[f];
            const float be = sBe[f];
            v8f Dk = {0.f, 0.f, 0.f, 0.f, 0.f, 0.f, 0.f, 0.f};
#pragma unroll
            for (int k0 = 0; k0 < 16; k0 += 4) {
                const int r0 = k0 + half * 2;             // this lane's K pair (n' rows)
                v2f a;                                    // A: P[M=l16][r0..r0+1]
                a.x = myP[l16 * 16 + r0];
                a.y = myP[l16 * 16 + r0 + 1];
                float x0 = x[xbase + (long)(n0 + r0) * D_ + f];
                float x1 = x[xbase + (long)(n0 + r0 + 1) * D_ + f];
                v2f bm;                                   // B: xn[K=r0..r0+1][N=f]
                bm.x = (x0 - myMu[r0])     * myRs[r0]     * g + be;
                bm.y = (x1 - myMu[r0 + 1]) * myRs[r0 + 1] * g + be;
                Dk = __builtin_amdgcn_wmma_f32_16x16x4_f32(false, a, false, bm,
                                                           (short)0, Dk, false, false);
            }
            if (half == 0) {                              // valid rows m=0..7 live in lanes 0-15
#pragma unroll
                for (int r = 0; r < 8; ++r) {
                    float old = myAcc[r * D_ + f];
                    myAcc[r * D_ + f] = old * corr[r] + Dk[r];
                }
            }
        }
    }

    // ---- publish per-wave softmax state, merge the 4 waves, emit chunk partial
    if (lane == 0) {
#pragma unroll
        for (int r = 0; r < 8; ++r) {
            sWmax[wave * 8 + r] = rowmax[r];
            sWl[wave * 8 + r]   = rowsum[r];
        }
    }
    __syncthreads();

    const int pidx = blockIdx.x;
    if (tid < 8) {
        float mb = NEG_BIG;
        for (int w = 0; w < NWAVES; ++w) mb = fmaxf(mb, sWmax[w * 8 + tid]);
        float lb = 0.0f;
        for (int w = 0; w < NWAVES; ++w)
            lb += sWl[w * 8 + tid] * __expf(sWmax[w * 8 + tid] - mb);
        sBmax[tid] = mb;
        pmax[pidx * 8 + tid] = mb;
        pl[pidx * 8 + tid]   = lb;
    }
    __syncthreads();

    for (int e = tid; e < 8 * D_; e += BLOCK) {
        const int m  = e / D_;
        const float mb = sBmax[m];
        float sum = 0.0f;
        for (int w = 0; w < NWAVES; ++w)
            sum += sAcc[w * 8 * D_ + e] * __expf(sWmax[w * 8 + m] - mb);
        pacc[(long)pidx * 8 * D_ + e] = sum;
    }
}

extern "C" __global__ __launch_bounds__(256)
void pm_reduce(const float* __restrict__ pmax, const float* __restrict__ pl,
               const float* __restrict__ pacc, float* __restrict__ out)
{
    __shared__ float gLinv[8], cw[CHUNKS * 8];
    const int b   = blockIdx.x;
    const int tid = threadIdx.x;
    if (tid < 8) {
        float mb = NEG_BIG;
        for (int c = 0; c < CHUNKS; ++c)
            mb = fmaxf(mb, pmax[(b * CHUNKS + c) * 8 + tid]);
        float lb = 0.0f;
        for (int c = 0; c < CHUNKS; ++c) {
            float w = __expf(pmax[(b * CHUNKS + c) * 8 + tid] - mb);
            cw[c * 8 + tid] = w;
            lb += pl[(b * CHUNKS + c) * 8 + tid] * w;
        }
        gLinv[tid] = 1.0f / lb;
    }
    __syncthreads();
    for (int e = tid; e < 8 * D_; e += 256) {
        const int m = e / D_;
        float s = 0.0f;
        for (int c = 0; c < CHUNKS; ++c)
            s += pacc[(long)(b * CHUNKS + c) * 8 * D_ + e] * cw[c * 8 + m];
        out[(long)b * 8 * D_ + e] = s * gLinv[m];
    }
}

extern "C" void kernel_launch(void* const* d_in, const int* in_sizes, int n_in,
                              void* d_out, int out_size, void* d_ws, size_t ws_size,
                              hipStream_t stream)
{
    const float* x     = (const float*)d_in[0];
    const float* q     = (const float*)d_in[1];
    const float* gamma = (const float*)d_in[2];
    const float* beta  = (const float*)d_in[3];
    float* out = (float*)d_out;

    const int P = B_ * CHUNKS;                 // 128 partials
    float* pmax = (float*)d_ws;                // [P][8]
    float* pl   = pmax + (size_t)P * 8;        // [P][8]
    float* pacc = pl + (size_t)P * 8;          // [P][8][768]  (~3.15 MB total)

    const size_t shmem = (size_t)LDS_FLOATS * sizeof(float);   // ~155 KB
    hipFuncSetAttribute(reinterpret_cast<const void*>(pm_fused),
                        hipFuncAttributeMaxDynamicSharedMemorySize, (int)shmem);

    pm_fused<<<P, BLOCK, shmem, stream>>>(x, q, gamma, beta, pmax, pl, pacc);
    pm_reduce<<<B_, 256, 0, stream>>>(pmax, pl, pacc, out);
}